// SelfSupervisedLoss_56916906606979
// MI455X (gfx1250) — compile-verified
//
#include <hip/hip_runtime.h>

// ---------------------------------------------------------------------------
// SelfSupervisedLoss for MI455X (gfx1250, wave32)
//
// a_energy[b,s] = sum_k A_val[k] * x[col[k],s] * x[row[k],s],  b = row[k]/N
// load[b,s]     = sum_n mass[b,n] * rhs[b,n,s] * x_hat[b,n,s]
// vol[b]        = sum_n mass[b,n]
// out = mean over (b,s) of ( w_kkt*kkt_e - w_comp*comp_e )
// ---------------------------------------------------------------------------

constexpr int   B_    = 4;
constexpr int   N_    = 100000;
constexpr int   NSOL_ = 8;
constexpr int   NNZ_  = 12800000;
constexpr float LAMB_COMP_ = 0.5f;

typedef __attribute__((ext_vector_type(2))) float v2f;
typedef __attribute__((ext_vector_type(8))) float v8f;

// ws layout (floats): [0,32) a_energy[b*8+s] | [32,64) load[b*8+s] | [64,68) vol[b]

__device__ __forceinline__ float waveReduceSum(float v) {
#pragma unroll
  for (int off = 16; off > 0; off >>= 1)
    v += __shfl_xor(v, off, 32);
  return v;
}

// ---------------------------------------------------------------------------
// Kernel 1: COO "energy" reduction. Streams rows/cols/vals (256 MB, coalesced),
// gathers x[row], x[col] as 2x float4 (L2-resident, 12.8 MB working set).
// Batch index is random per lane -> use 4 masked FMAs instead of branching.
// ---------------------------------------------------------------------------
__global__ void __launch_bounds__(256)
spmv_energy_kernel(const long long* __restrict__ rows,
                   const long long* __restrict__ cols,
                   const float* __restrict__ vals,
                   const float* __restrict__ x,     // [B*N, 8] flat
                   float* __restrict__ ws) {
  float acc[B_][NSOL_];
#pragma unroll
  for (int b = 0; b < B_; ++b)
#pragma unroll
    for (int s = 0; s < NSOL_; ++s) acc[b][s] = 0.f;

  const int stride = gridDim.x * blockDim.x;
  for (int k = blockIdx.x * blockDim.x + threadIdx.x; k < NNZ_; k += stride) {
    const long long r = rows[k];
    const long long c = cols[k];
    const float     v = vals[k];
    const unsigned  b = (unsigned)r / (unsigned)N_;   // magic-number div

    const float4* xr = reinterpret_cast<const float4*>(x + ((size_t)r << 3));
    const float4* xc = reinterpret_cast<const float4*>(x + ((size_t)c << 3));
    const float4 r0 = xr[0], r1 = xr[1];
    const float4 c0 = xc[0], c1 = xc[1];

    const float t[NSOL_] = {r0.x * c0.x, r0.y * c0.y, r0.z * c0.z, r0.w * c0.w,
                            r1.x * c1.x, r1.y * c1.y, r1.z * c1.z, r1.w * c1.w};
    const float m[B_] = {b == 0u ? v : 0.f, b == 1u ? v : 0.f,
                         b == 2u ? v : 0.f, b == 3u ? v : 0.f};
#pragma unroll
    for (int bb = 0; bb < B_; ++bb)
#pragma unroll
      for (int s = 0; s < NSOL_; ++s)
        acc[bb][s] = fmaf(m[bb], t[s], acc[bb][s]);
  }

  // wave32 butterfly -> LDS combine -> 32 global f32 atomics per block
  __shared__ float sacc[32];
  if (threadIdx.x < 32) sacc[threadIdx.x] = 0.f;
  __syncthreads();
#pragma unroll
  for (int b = 0; b < B_; ++b)
#pragma unroll
    for (int s = 0; s < NSOL_; ++s) {
      const float v = waveReduceSum(acc[b][s]);
      if ((threadIdx.x & 31) == 0) atomicAdd(&sacc[b * NSOL_ + s], v);
    }
  __syncthreads();
  if (threadIdx.x < 32) atomicAdd(&ws[threadIdx.x], sacc[threadIdx.x]);
}

// ---------------------------------------------------------------------------
// Kernel 2: load[b,s] and vol[b]. One batch per blockIdx.y -> each thread keeps
// only 9 accumulators. Pure streaming (27 MB total).
// ---------------------------------------------------------------------------
__global__ void __launch_bounds__(256)
load_vol_kernel(const float* __restrict__ x_hat,
                const float* __restrict__ rhs,
                const float* __restrict__ mass,
                float* __restrict__ ws) {
  const int b = blockIdx.y;
  float al[NSOL_] = {0.f, 0.f, 0.f, 0.f, 0.f, 0.f, 0.f, 0.f};
  float av = 0.f;

  const int stride = gridDim.x * blockDim.x;
  for (int n = blockIdx.x * blockDim.x + threadIdx.x; n < N_; n += stride) {
    const size_t row = (size_t)b * N_ + n;
    const float  m   = mass[row];
    const float4* rp = reinterpret_cast<const float4*>(rhs   + (row << 3));
    const float4* xp = reinterpret_cast<const float4*>(x_hat + (row << 3));
    const float4 r0 = rp[0], r1 = rp[1];
    const float4 x0 = xp[0], x1 = xp[1];
    av += m;
    al[0] = fmaf(m * r0.x, x0.x, al[0]);
    al[1] = fmaf(m * r0.y, x0.y, al[1]);
    al[2] = fmaf(m * r0.z, x0.z, al[2]);
    al[3] = fmaf(m * r0.w, x0.w, al[3]);
    al[4] = fmaf(m * r1.x, x1.x, al[4]);
    al[5] = fmaf(m * r1.y, x1.y, al[5]);
    al[6] = fmaf(m * r1.z, x1.z, al[6]);
    al[7] = fmaf(m * r1.w, x1.w, al[7]);
  }

  __shared__ float sacc[9];
  if (threadIdx.x < 9) sacc[threadIdx.x] = 0.f;
  __syncthreads();
#pragma unroll
  for (int s = 0; s < NSOL_; ++s) {
    const float v = waveReduceSum(al[s]);
    if ((threadIdx.x & 31) == 0) atomicAdd(&sacc[s], v);
  }
  {
    const float v = waveReduceSum(av);
    if ((threadIdx.x & 31) == 0) atomicAdd(&sacc[8], v);
  }
  __syncthreads();
  if (threadIdx.x < 8) atomicAdd(&ws[32 + b * NSOL_ + threadIdx.x], sacc[threadIdx.x]);
  if (threadIdx.x == 8) atomicAdd(&ws[64 + b], sacc[8]);
}

// ---------------------------------------------------------------------------
// Kernel 3: finalize. One wave32; lane (b,s) = (lane>>3, lane&7) computes its
// per-element term, then the 32-lane mean is taken with V_WMMA_F32_16X16X4_F32:
//   A[m][0] = t[lane m] (lanes 0-15), A[m][2] = t[lane m+16], B = ones(4x16)
//   => D[m][n] = t[m] + t[m+16];  sum of a lane's 8 C/D VGPRs + shfl_xor(16)
//   = full 32-lane sum. EXEC is all-ones (32 threads, no divergence).
// ---------------------------------------------------------------------------
__global__ void __launch_bounds__(32)
finalize_kernel(const float* __restrict__ ws, float* __restrict__ out) {
  const int lane = threadIdx.x;
  const int b    = lane >> 3;

  const float ae  = ws[lane];
  const float ld  = ws[32 + lane];
  const float vol = ws[64 + b];

  const float sigma  = ld / fmaxf(ae, 1e-4f);
  const float kkt_e  = (0.5f * ae * sigma - ld) * sigma / vol;
  const float comp_e = sigma * ld / vol;
  const float w_kkt  = 1.0f - LAMB_COMP_;
  const float t      = w_kkt * kkt_e - LAMB_COMP_ * comp_e;

  v2f a;  a[0] = t;   a[1] = 0.f;   // K0 (lanes 0-15) / K2 (lanes 16-31) = t
  v2f bm; bm[0] = 1.f; bm[1] = 1.f; // B = all ones
  v8f cc = {};
  cc = __builtin_amdgcn_wmma_f32_16x16x4_f32(
      /*neg_a=*/false, a, /*neg_b=*/false, bm,
      /*c_mod=*/(short)0, cc, /*reuse_a=*/false, /*reuse_b=*/false);

  float local = cc[0] + cc[1] + cc[2] + cc[3] + cc[4] + cc[5] + cc[6] + cc[7];
  const float total = local + __shfl_xor(local, 16, 32);
  if (lane == 0) out[0] = total * (1.0f / 32.0f);
}

// ---------------------------------------------------------------------------
extern "C" void kernel_launch(void* const* d_in, const int* in_sizes, int n_in,
                              void* d_out, int out_size, void* d_ws, size_t ws_size,
                              hipStream_t stream) {
  const float*     x_hat = (const float*)d_in[0];
  const float*     rhs   = (const float*)d_in[1];
  const long long* A_ind = (const long long*)d_in[2];   // [2, NNZ] int64
  const float*     A_val = (const float*)d_in[3];
  // d_in[4] (subspace_vectors) is unused by the reference
  const float*     mass  = (const float*)d_in[5];
  float*           ws    = (float*)d_ws;

  // zero the 68-float accumulator region every call (graph-capture safe)
  hipMemsetAsync(d_ws, 0, 68 * sizeof(float), stream);

  spmv_energy_kernel<<<1024, 256, 0, stream>>>(A_ind, A_ind + NNZ_, A_val, x_hat, ws);
  load_vol_kernel<<<dim3(64, B_), 256, 0, stream>>>(x_hat, rhs, mass, ws);
  finalize_kernel<<<1, 32, 0, stream>>>(ws, (float*)d_out);
}